// NeuronBank_300647710818
// MI455X (gfx1250) — compile-verified
//
#include <hip/hip_runtime.h>
#include <hip/hip_bf16.h>

// Problem constants (from reference):
//   N=2048, D=256, R=32, B=2, S=512, K=4
//   W_Q/W_K/W_V: [N, D, R] fp32 ; W_O: [N, R, D] fp32 ; indices: [B,S,K] int32
//   Outputs (concat flat): WQ_sel, WK_sel, WV_sel [B,S,K,D,R], WO_sel [B,S,K,R,D], loss (1)
#define NB_N     2048
#define NB_D     256
#define NB_R     32
#define NB_GATH  4096                 // B*S*K
#define NB_BLKF  8192                 // D*R floats per neuron block
#define NB_TENSZ 33554432ull          // NB_GATH * NB_BLKF
#define NB_LOSS_OFF 134217728ull      // 4 * NB_TENSZ

typedef __attribute__((ext_vector_type(2))) float v2f;
typedef __attribute__((ext_vector_type(4))) float v4f;
typedef __attribute__((ext_vector_type(8))) float v8f;

// ---------------------------------------------------------------------------
// Kernel 1: gather copy.  One block per (tensor, gather-entry); each block
// copies one contiguous 32KB neuron block with b128 loads + non-temporal
// b128 stores (output is write-once, keep it out of L2).
// ---------------------------------------------------------------------------
__global__ void __launch_bounds__(256)
nb_gather_kernel(const int* __restrict__ idx,
                 const float* __restrict__ Wq, const float* __restrict__ Wk,
                 const float* __restrict__ Wv, const float* __restrict__ Wo,
                 float* __restrict__ out)
{
    const int t = blockIdx.x >> 12;      // tensor 0..3
    const int g = blockIdx.x & 4095;     // gather entry 0..4095
    const float* W = (t == 0) ? Wq : (t == 1) ? Wk : (t == 2) ? Wv : Wo;
    const int n = idx[g];                // uniform across block -> scalar load

    const v4f* __restrict__ src = (const v4f*)(W + (size_t)n * NB_BLKF);
    v4f* __restrict__ dst = (v4f*)(out + (size_t)t * NB_TENSZ + (size_t)g * NB_BLKF);

    // 8192 floats = 2048 float4 ; 256 threads x 8 float4 each
#pragma unroll
    for (int i = 0; i < 8; ++i) {
        const int j = i * 256 + (int)threadIdx.x;
        v4f v = src[j];
        __builtin_nontemporal_store(v, &dst[j]);
    }
}

// ---------------------------------------------------------------------------
// Kernel 2: ortho loss via V_WMMA_F32_16X16X4_F32.
// One wave computes WtW (32x32) for one (tensor, neuron) pair.
// For this A^T*A shape the A(16x4) and B(4x16) per-lane layouts coincide:
//   lane L, vgpr v  ->  element (d = d0 + 2*(L>=16) + v, col = L%16 + 16*tile)
// Symmetry: compute tiles (0,0), (1,1) and (0,1); (1,0) contributes 2x of (0,1).
// SD = stride of d-index, SC = stride of column index in memory.
// ---------------------------------------------------------------------------
template <int SD, int SC>
__device__ __forceinline__ float nb_wtw_loss_partial(const float* __restrict__ base)
{
    const int lane = (int)(threadIdx.x & 31u);
    const int half = lane >> 4;          // 0: K=0..1 ; 1: K=2..3
    const int l    = lane & 15;
    const int c0   = l;                  // columns 0..15
    const int c1   = l + 16;             // columns 16..31

    v8f acc00 = {}; v8f acc01 = {}; v8f acc11 = {};

#pragma unroll 4
    for (int d0 = 0; d0 < NB_D; d0 += 4) {
        const int dlo = d0 + 2 * half;
        v2f f0, f1;
        f0.x = base[(size_t)dlo * SD + (size_t)c0 * SC];
        f0.y = base[(size_t)(dlo + 1) * SD + (size_t)c0 * SC];
        f1.x = base[(size_t)dlo * SD + (size_t)c1 * SC];
        f1.y = base[(size_t)(dlo + 1) * SD + (size_t)c1 * SC];

        acc00 = __builtin_amdgcn_wmma_f32_16x16x4_f32(false, f0, false, f0,
                                                      (short)0, acc00, false, false);
        acc01 = __builtin_amdgcn_wmma_f32_16x16x4_f32(false, f0, false, f1,
                                                      (short)0, acc01, false, false);
        acc11 = __builtin_amdgcn_wmma_f32_16x16x4_f32(false, f1, false, f1,
                                                      (short)0, acc11, false, false);
    }

    // C/D layout: lane L, vgpr v -> (Mloc = 8*(L>=16)+v, Nloc = L%16)
    float sum = 0.0f;
#pragma unroll
    for (int v = 0; v < 8; ++v) {
        const float ident = ((half * 8 + v) == l) ? 1.0f : 0.0f;
        const float e00 = acc00[v] - ident;   // tile (0,0): diag when Mloc==Nloc
        const float e11 = acc11[v] - ident;   // tile (1,1): diag when Mloc==Nloc
        const float e01 = acc01[v];           // tile (0,1): never diagonal
        sum += e00 * e00 + e11 * e11 + 2.0f * e01 * e01;
    }
    return sum;
}

__global__ void __launch_bounds__(256)
nb_wtw_loss_kernel(const float* __restrict__ Wq, const float* __restrict__ Wk,
                   const float* __restrict__ Wv, const float* __restrict__ Wo,
                   float* __restrict__ acc)
{
    const int w = (int)blockIdx.x * 8 + (int)(threadIdx.x >> 5);  // global wave id
    const int t = w >> 11;        // tensor 0..3
    const int n = w & (NB_N - 1); // neuron 0..2047

    float sum;
    if (t < 3) {
        const float* W = (t == 0) ? Wq : (t == 1) ? Wk : Wv;
        const float* base = W + (size_t)n * NB_BLKF;     // [D,R]: d-stride R, col-stride 1
        sum = nb_wtw_loss_partial<NB_R, 1>(base);
    } else {
        const float* base = Wo + (size_t)n * NB_BLKF;    // [R,D]: d-stride 1, col-stride D
        sum = nb_wtw_loss_partial<1, NB_D>(base);
    }

    // wave32 tree reduction
#pragma unroll
    for (int off = 16; off > 0; off >>= 1)
        sum += __shfl_down(sum, off, 32);

    if ((threadIdx.x & 31u) == 0u)
        atomicAdd(acc, sum);
}

// ---------------------------------------------------------------------------
// Workspace init + finalize
// ---------------------------------------------------------------------------
__global__ void nb_init_ws(float* acc) { *acc = 0.0f; }

__global__ void nb_finalize(const float* __restrict__ acc, float* __restrict__ out_loss)
{
    // loss = raw_sum / (N*R*R) / 4  =  raw_sum / 8388608
    *out_loss = *acc * (1.0f / 8388608.0f);
}

// ---------------------------------------------------------------------------
extern "C" void kernel_launch(void* const* d_in, const int* in_sizes, int n_in,
                              void* d_out, int out_size, void* d_ws, size_t ws_size,
                              hipStream_t stream)
{
    const int*   idx = (const int*)d_in[0];
    const float* Wq  = (const float*)d_in[1];
    const float* Wk  = (const float*)d_in[2];
    const float* Wv  = (const float*)d_in[3];
    const float* Wo  = (const float*)d_in[4];
    float* out = (float*)d_out;
    float* acc = (float*)d_ws;

    nb_init_ws<<<1, 1, 0, stream>>>(acc);

    // 4 tensors * 4096 gathers = 16384 blocks, 32KB copied per block
    nb_gather_kernel<<<16384, 256, 0, stream>>>(idx, Wq, Wk, Wv, Wo, out);

    // 8192 (tensor, neuron) waves, 8 waves per block
    nb_wtw_loss_kernel<<<1024, 256, 0, stream>>>(Wq, Wk, Wv, Wo, acc);

    nb_finalize<<<1, 1, 0, stream>>>(acc, out + NB_LOSS_OFF);
}